// Sequence2_14104672600668
// MI455X (gfx1250) — compile-verified
//
#include <hip/hip_runtime.h>
#include <hip/hip_bf16.h>

#define BSZ 256
#define TLEN 512
#define INDIM 128
#define HID 512
#define OUTDIM 128

typedef __attribute__((ext_vector_type(2))) float v2f;
typedef __attribute__((ext_vector_type(8))) float v8f;

// ---------------------------------------------------------------------------
// GEMM tile accumulator (fp32 WMMA 16x16x4).
// Block = 256 threads = 8 waves; block tile = 64 rows x 16 h-columns x 4 gates.
// Wave wv: gate g = wv&3, row-group rg = wv>>2 (32 rows); each wave keeps TWO
// 16x16 accumulators (row halves) and reuses each B fragment for both -> B LDS
// traffic per WMMA halved, two independent accumulation chains for pipelining.
// Fragments for a full 32-wide K chunk are pre-loaded into VGPR arrays so the
// ds_loads clause together and the 16 WMMAs issue as a dense burst.
// ---------------------------------------------------------------------------
__device__ __forceinline__ void gemm_tile_accum(
    const float* __restrict__ A, int lda, int Kdim,
    const float* __restrict__ W, int ldw,
    int m0, int n0,
    float (&As)[64][33], float (&Bs)[4][32][17],
    v8f& acc0, v8f& acc1)
{
    const int tid  = threadIdx.x;
    const int lane = tid & 31;
    const int wv   = tid >> 5;
    const int g    = wv & 3;
    const int rb   = (wv >> 2) << 5;     // wave row base: 0 or 32
    const int mm   = lane & 15;
    const int kh   = (lane >> 4) << 1;   // 0 or 2

    // A loader: 64 rows x 32 k, 8 contiguous floats per thread
    const int arow = tid >> 2;           // 0..63
    const int akq  = (tid & 3) << 3;     // 0,8,16,24
    // B loader: 4 gates x 16 n x 32 k, 8 contiguous floats per thread
    const int bg = tid >> 6;             // 0..3
    const int bt = tid & 63;
    const int bn = bt >> 2;              // 0..15
    const int bk = (bt & 3) << 3;        // 0,8,16,24

    for (int kb = 0; kb < Kdim; kb += 32) {
        __syncthreads();
        {
            const float* asrc = A + (size_t)(m0 + arow) * lda + kb + akq;
            const float4 a0 = *(const float4*)(asrc);
            const float4 a1 = *(const float4*)(asrc + 4);
            As[arow][akq + 0] = a0.x;
            As[arow][akq + 1] = a0.y;
            As[arow][akq + 2] = a0.z;
            As[arow][akq + 3] = a0.w;
            As[arow][akq + 4] = a1.x;
            As[arow][akq + 5] = a1.y;
            As[arow][akq + 6] = a1.z;
            As[arow][akq + 7] = a1.w;

            const float* wsrc = W + (size_t)(bg * HID + n0 + bn) * ldw + kb + bk;
            const float4 w0 = *(const float4*)(wsrc);
            const float4 w1 = *(const float4*)(wsrc + 4);
            Bs[bg][bk + 0][bn] = w0.x;
            Bs[bg][bk + 1][bn] = w0.y;
            Bs[bg][bk + 2][bn] = w0.z;
            Bs[bg][bk + 3][bn] = w0.w;
            Bs[bg][bk + 4][bn] = w1.x;
            Bs[bg][bk + 5][bn] = w1.y;
            Bs[bg][bk + 6][bn] = w1.z;
            Bs[bg][bk + 7][bn] = w1.w;
        }
        __syncthreads();

        // Pre-load all fragments for this chunk into registers.
        v2f af0[8], af1[8], bf[8];
        #pragma unroll
        for (int kk = 0; kk < 8; ++kk) {
            const int k = kk << 2;
            // A 16x4: lane = M, VGPR v holds K = v + 2*(lane>>4)
            af0[kk].x = As[rb + mm][k + kh];
            af0[kk].y = As[rb + mm][k + kh + 1];
            af1[kk].x = As[rb + 16 + mm][k + kh];
            af1[kk].y = As[rb + 16 + mm][k + kh + 1];
            // B 4x16: VGPR v holds K-row = v + 2*(lane>>4), N = lane&15
            bf[kk].x = Bs[g][k + kh][mm];
            bf[kk].y = Bs[g][k + kh + 1][mm];
        }
        #pragma unroll
        for (int kk = 0; kk < 8; ++kk) {
            acc0 = __builtin_amdgcn_wmma_f32_16x16x4_f32(
                false, af0[kk], false, bf[kk], (short)0, acc0, false, false);
            acc1 = __builtin_amdgcn_wmma_f32_16x16x4_f32(
                false, af1[kk], false, bf[kk], (short)0, acc1, false, false);
        }
    }
}

// ---------------------------------------------------------------------------
// Fused LSTM layer step: gates = X@W_ih^T + Hprev@W_hh^T + b_ih + b_hh,
// then i/f/g/o activations and c,h update. Block tile: 64 rows x 16 h-cols,
// all 4 gates resident in-block (exchanged via LDS).
// Grid: (BSZ/64, HID/16) = (4, 32).
// ---------------------------------------------------------------------------
__global__ __launch_bounds__(256) void lstm_layer_kernel(
    const float* __restrict__ X, int ldx, int Kx,
    const float* __restrict__ Hprev,
    const float* __restrict__ W_ih, const float* __restrict__ W_hh,
    const float* __restrict__ b_ih, const float* __restrict__ b_hh,
    float* __restrict__ Hout, float* __restrict__ C)
{
    __shared__ float As[64][33];
    __shared__ float Bs[4][32][17];
    __shared__ float Gx[4][64][17];

    const int tid  = threadIdx.x;
    const int lane = tid & 31;
    const int wv   = tid >> 5;
    const int g    = wv & 3;
    const int rb   = (wv >> 2) << 5;
    const int m0   = blockIdx.x << 6;   // row block (64)
    const int n0   = blockIdx.y << 4;   // h-column block (16)
    const int nn   = lane & 15;

    v8f acc0 = {};
    v8f acc1 = {};
    gemm_tile_accum(X, ldx, Kx, W_ih, Kx, m0, n0, As, Bs, acc0, acc1);
    gemm_tile_accum(Hprev, HID, HID, W_hh, HID, m0, n0, As, Bs, acc0, acc1);

    // bias depends only on column -> one scalar per lane
    const float bias = b_ih[g * HID + n0 + nn] + b_hh[g * HID + n0 + nn];
    #pragma unroll
    for (int v = 0; v < 8; ++v) { acc0[v] += bias; acc1[v] += bias; }

    // write gate tiles to LDS for cross-wave exchange
    // (C/D layout: M = v + 8*(lane>>4))
    #pragma unroll
    for (int v = 0; v < 8; ++v) {
        const int m = v + ((lane >> 4) << 3);
        Gx[g][rb + m][nn]      = acc0[v];
        Gx[g][rb + 16 + m][nn] = acc1[v];
    }
    __syncthreads();

    // pointwise LSTM cell: 64 rows x 16 cols = 1024 elems, 4 per thread
    #pragma unroll
    for (int e = 0; e < 4; ++e) {
        const int idx = tid + (e << 8);
        const int ml  = idx >> 4;        // 0..63
        const int nc  = idx & 15;
        const float gi = Gx[0][ml][nc];
        const float gf = Gx[1][ml][nc];
        const float gg = Gx[2][ml][nc];
        const float go = Gx[3][ml][nc];
        const float si = 1.0f / (1.0f + __expf(-gi));
        const float sf = 1.0f / (1.0f + __expf(-gf));
        const float tg = tanhf(gg);
        const float so = 1.0f / (1.0f + __expf(-go));
        const size_t off = (size_t)(m0 + ml) * HID + n0 + nc;
        const float cn = sf * C[off] + si * tg;
        C[off]    = cn;
        Hout[off] = so * tanhf(cn);
    }
}

// ---------------------------------------------------------------------------
// Projection: out[:, t, :] = H2 @ W_lin^T + b_lin.  Block tile 32x64 (8 waves,
// 2 row-halves x 4 col-groups). Grid: (BSZ/32, OUTDIM/64) = (8, 2).
// ---------------------------------------------------------------------------
__global__ __launch_bounds__(256) void proj_kernel(
    const float* __restrict__ H2,
    const float* __restrict__ W, const float* __restrict__ bias,
    float* __restrict__ Out, int ldo)
{
    __shared__ float As[32][33];
    __shared__ float Bs[32][65];

    const int tid  = threadIdx.x;
    const int lane = tid & 31;
    const int wv   = tid >> 5;
    const int cw   = wv & 3;
    const int m0w  = (wv >> 2) << 4;
    const int m0   = blockIdx.x << 5;
    const int n0   = blockIdx.y << 6;
    const int mm   = lane & 15;
    const int kh   = (lane >> 4) << 1;

    const int arow = tid >> 3;
    const int akq  = (tid & 7) << 2;
    const int bn   = tid >> 2;          // 0..63
    const int bk   = (tid & 3) << 3;    // 0,8,16,24

    v8f acc = {};
    for (int kb = 0; kb < HID; kb += 32) {
        __syncthreads();
        const float4 av = *(const float4*)(H2 + (size_t)(m0 + arow) * HID + kb + akq);
        As[arow][akq + 0] = av.x;
        As[arow][akq + 1] = av.y;
        As[arow][akq + 2] = av.z;
        As[arow][akq + 3] = av.w;
        const float* wsrc = W + (size_t)(n0 + bn) * HID + kb + bk;
        const float4 w0 = *(const float4*)wsrc;
        const float4 w1 = *(const float4*)(wsrc + 4);
        Bs[bk + 0][bn] = w0.x;
        Bs[bk + 1][bn] = w0.y;
        Bs[bk + 2][bn] = w0.z;
        Bs[bk + 3][bn] = w0.w;
        Bs[bk + 4][bn] = w1.x;
        Bs[bk + 5][bn] = w1.y;
        Bs[bk + 6][bn] = w1.z;
        Bs[bk + 7][bn] = w1.w;
        __syncthreads();

        v2f af[8], bf[8];
        #pragma unroll
        for (int kk = 0; kk < 8; ++kk) {
            const int k = kk << 2;
            af[kk].x = As[m0w + mm][k + kh];
            af[kk].y = As[m0w + mm][k + kh + 1];
            bf[kk].x = Bs[k + kh][(cw << 4) + mm];
            bf[kk].y = Bs[k + kh + 1][(cw << 4) + mm];
        }
        #pragma unroll
        for (int kk = 0; kk < 8; ++kk) {
            acc = __builtin_amdgcn_wmma_f32_16x16x4_f32(
                false, af[kk], false, bf[kk], (short)0, acc, false, false);
        }
    }

    const int ng = n0 + (cw << 4) + mm;
    const float bv = bias[ng];
    #pragma unroll
    for (int v = 0; v < 8; ++v) {
        const int m = m0 + m0w + v + ((lane >> 4) << 3);
        Out[(size_t)m * ldo + ng] = acc[v] + bv;
    }
}

__global__ void zero_kernel(float* __restrict__ p, int n) {
    const int i = blockIdx.x * blockDim.x + threadIdx.x;
    if (i < n) p[i] = 0.0f;
}

// ---------------------------------------------------------------------------
extern "C" void kernel_launch(void* const* d_in, const int* in_sizes, int n_in,
                              void* d_out, int out_size, void* d_ws, size_t ws_size,
                              hipStream_t stream)
{
    const float* input = (const float*)d_in[0];
    // d_in[1] = future (scalar int, == 0 in this workload)
    const float* W_ih1 = (const float*)d_in[2];
    const float* W_hh1 = (const float*)d_in[3];
    const float* b_ih1 = (const float*)d_in[4];
    const float* b_hh1 = (const float*)d_in[5];
    const float* W_ih2 = (const float*)d_in[6];
    const float* W_hh2 = (const float*)d_in[7];
    const float* b_ih2 = (const float*)d_in[8];
    const float* b_hh2 = (const float*)d_in[9];
    const float* W_lin = (const float*)d_in[10];
    const float* b_lin = (const float*)d_in[11];
    float* out = (float*)d_out;

    const size_t S = (size_t)BSZ * HID;   // 131072 floats per state buffer
    float* ws = (float*)d_ws;
    float* h1[2] = {ws + 0 * S, ws + 1 * S};
    float* c1    =  ws + 2 * S;
    float* h2[2] = {ws + 3 * S, ws + 4 * S};
    float* c2    =  ws + 5 * S;

    const int zn = (int)(6 * S);
    zero_kernel<<<(zn + 255) / 256, 256, 0, stream>>>(ws, zn);

    const dim3 blk(256);
    const dim3 gl(BSZ / 64, HID / 16);      // (4, 32)
    const dim3 gp(BSZ / 32, OUTDIM / 64);   // (8, 2)

    for (int t = 0; t < TLEN; ++t) {
        const float* x = input + (size_t)t * INDIM;   // rows stride T*INDIM
        lstm_layer_kernel<<<gl, blk, 0, stream>>>(
            x, TLEN * INDIM, INDIM,
            h1[t & 1], W_ih1, W_hh1, b_ih1, b_hh1,
            h1[(t + 1) & 1], c1);
        lstm_layer_kernel<<<gl, blk, 0, stream>>>(
            h1[(t + 1) & 1], HID, HID,
            h2[t & 1], W_ih2, W_hh2, b_ih2, b_hh2,
            h2[(t + 1) & 1], c2);
        proj_kernel<<<gp, blk, 0, stream>>>(
            h2[(t + 1) & 1], W_lin, b_lin,
            out + (size_t)t * OUTDIM, TLEN * OUTDIM);
    }
}